// FinalCoarseToFineIGAModule_4045859192854
// MI455X (gfx1250) — compile-verified
//
#include <hip/hip_runtime.h>

#define BB 2
#define NN 4096
#define KK 512
#define CC 384
#define RR 16
#define JITTER_ 1e-4f
#define NEG_BIG -1.0e9f
#define LDA 386   // padded LDS pitch (floats) to avoid 64-bank conflicts

typedef __attribute__((ext_vector_type(2))) float v2f;
typedef __attribute__((ext_vector_type(8))) float v8f;

__device__ __forceinline__ float wave_sum32(float v) {
#pragma unroll
  for (int off = 16; off > 0; off >>= 1) v += __shfl_xor(v, off, 32);
  return v;
}

// ---------------------------------------------------------------- K0: mask sums
__global__ void __launch_bounds__(256) k0_mask_sums(const float* __restrict__ node_mask,
                                                    float* __restrict__ scal) {
  __shared__ float red[256];
  int tid = threadIdx.x;
  float s0 = 0.f, s1 = 0.f;
  for (int i = tid; i < NN; i += 256) { s0 += node_mask[i]; s1 += node_mask[NN + i]; }
  red[tid] = s0; __syncthreads();
  for (int off = 128; off > 0; off >>= 1) { if (tid < off) red[tid] += red[tid + off]; __syncthreads(); }
  float S0 = red[0]; __syncthreads();
  red[tid] = s1; __syncthreads();
  for (int off = 128; off > 0; off >>= 1) { if (tid < off) red[tid] += red[tid + off]; __syncthreads(); }
  float S1 = red[0];
  if (tid == 0) {
    scal[0] = fmaxf(fmaxf(S0, 1.f) - 1.f, 1.f);  // denom b=0
    scal[1] = fmaxf(fmaxf(S1, 1.f) - 1.f, 1.f);  // denom b=1
    scal[2] = S0 + S1;                           // total mask sum
  }
}

// ---------------------------------------------------- K1: Fourier embed + LayerNorm -> q0
__global__ void __launch_bounds__(256) k1_embed_ln(const int* __restrict__ res_idx,
                                                   const float* __restrict__ node_mask,
                                                   const float* __restrict__ Bmat,
                                                   const float* __restrict__ ln_g,
                                                   const float* __restrict__ ln_b,
                                                   const float* __restrict__ scal,
                                                   float* __restrict__ q0) {
  int wave = threadIdx.x >> 5, lane = threadIdx.x & 31;
  int node = blockIdx.x * 8 + wave;           // flat b*N+n
  int b = node >> 12;
  float m = node_mask[node];
  int ri = res_idx[node];
  ri = ri < 0 ? 0 : (ri > 4095 ? 4095 : ri);
  float pos = fminf(fmaxf((float)ri / scal[b], 0.f), 1.f);
  const float TWO_PI = 6.28318530717958647692f;
  float e[12];
#pragma unroll
  for (int i = 0; i < 12; ++i) {
    int c = lane + 32 * i;
    float pr = (c < 192) ? cosf(TWO_PI * pos * Bmat[c]) : sinf(TWO_PI * pos * Bmat[c - 192]);
    e[i] = pr * m;
  }
  float s = 0.f;
#pragma unroll
  for (int i = 0; i < 12; ++i) s += e[i];
  float mean = wave_sum32(s) * (1.f / (float)CC);
  float v = 0.f;
#pragma unroll
  for (int i = 0; i < 12; ++i) { float d = e[i] - mean; v += d * d; }
  float rstd = rsqrtf(wave_sum32(v) * (1.f / (float)CC) + 1e-5f);
#pragma unroll
  for (int i = 0; i < 12; ++i) {
    int c = lane + 32 * i;
    q0[(size_t)node * CC + c] = (e[i] - mean) * rstd * ln_g[c] + ln_b[c];
  }
}

// -------------------------------------------- K2: WMMA GEMM  C[M,384] = A[M,384] * W[384,384]
__global__ void __launch_bounds__(128) k2_gemm_wmma(const float* __restrict__ A,
                                                    const float* __restrict__ W,
                                                    float* __restrict__ Cout) {
  __shared__ float as[16 * LDA];
  int tid = threadIdx.x, wave = tid >> 5, lane = tid & 31;
  int rbase = blockIdx.x * 16;
  int col   = blockIdx.y * 64 + wave * 16 + (lane & 15);
  // stage 16x384 A tile
  for (int idx = tid; idx < 16 * 96; idx += 128) {
    int row = idx / 96, c4 = (idx % 96) * 4;
    float4 val = *(const float4*)(A + (size_t)(rbase + row) * CC + c4);
    float* dst = &as[row * LDA + c4];
    dst[0] = val.x; dst[1] = val.y; dst[2] = val.z; dst[3] = val.w;
  }
  __syncthreads();
  v8f acc = {};
  int mrow = lane & 15;
  int koff = (lane >> 4) << 1;   // lanes 0-15: K+0/1, lanes 16-31: K+2/3 (ISA A layout)
#pragma unroll 8
  for (int kk = 0; kk < CC; kk += 4) {
    int kl = kk + koff;
    v2f a = *(const v2f*)&as[mrow * LDA + kl];
    v2f bf;
    bf.x = W[(size_t)kl * CC + col];
    bf.y = W[(size_t)(kl + 1) * CC + col];
    acc = __builtin_amdgcn_wmma_f32_16x16x4_f32(false, a, false, bf, (short)0, acc, false, false);
  }
  int half = lane >> 4;
#pragma unroll
  for (int r = 0; r < 8; ++r)
    Cout[(size_t)(rbase + r + 8 * half) * CC + col] = acc[r];   // C layout: VGPR r -> rows r, r+8
}

// ----------------------------- K3: fused WMMA logits (q . k^T) + masked argmax -> j0
__global__ void __launch_bounds__(128) k3_logits_argmax(const float* __restrict__ q,
                                                        const float* __restrict__ kmat,
                                                        const float* __restrict__ mask_parent,
                                                        int* __restrict__ j0) {
  __shared__ float qs[16 * LDA];
  __shared__ float redv[16][64];
  __shared__ int   redi[16][64];
  int tid = threadIdx.x, wave = tid >> 5, lane = tid & 31;
  int b = blockIdx.x / (NN / 16);
  int nbase = (blockIdx.x % (NN / 16)) * 16;
  const float* qbase = q + ((size_t)b * NN + nbase) * CC;
  for (int idx = tid; idx < 16 * 96; idx += 128) {
    int row = idx / 96, c4 = (idx % 96) * 4;
    float4 val = *(const float4*)(qbase + (size_t)row * CC + c4);
    float* dst = &qs[row * LDA + c4];
    dst[0] = val.x; dst[1] = val.y; dst[2] = val.z; dst[3] = val.w;
  }
  __syncthreads();
  const float inv_sqrt_c = 0.05103103630798288f;   // 1/sqrt(384)
  int mrow = lane & 15;
  int koff = (lane >> 4) << 1;
  float bestv[8]; int besti[8];
#pragma unroll
  for (int r = 0; r < 8; ++r) { bestv[r] = -__builtin_inff(); besti[r] = 0x7fffffff; }
  for (int t = 0; t < 8; ++t) {           // each wave covers 8 parent tiles of 16
    int p = (t * 4 + wave) * 16 + (lane & 15);
    const float* krow = kmat + ((size_t)b * KK + p) * CC;
    v8f acc = {};
#pragma unroll 8
    for (int kk = 0; kk < CC; kk += 4) {
      int kl = kk + koff;
      v2f a = *(const v2f*)&qs[mrow * LDA + kl];
      v2f bf; bf.x = krow[kl]; bf.y = krow[kl + 1];
      acc = __builtin_amdgcn_wmma_f32_16x16x4_f32(false, a, false, bf, (short)0, acc, false, false);
    }
    float bias = (mask_parent[(size_t)b * KK + p] - 1.0f) * 1.0e9f;
#pragma unroll
    for (int r = 0; r < 8; ++r) {
      float v = acc[r] * inv_sqrt_c + bias;
      if (v > bestv[r] || (v == bestv[r] && p < besti[r])) { bestv[r] = v; besti[r] = p; }
    }
  }
  int half = lane >> 4, colc = wave * 16 + (lane & 15);
#pragma unroll
  for (int r = 0; r < 8; ++r) {
    int nodeR = r + 8 * half;
    redv[nodeR][colc] = bestv[r];
    redi[nodeR][colc] = besti[r];
  }
  __syncthreads();
  if (tid < 16) {
    float bv = redv[tid][0]; int bi = redi[tid][0];
    for (int c = 1; c < 64; ++c) {
      float v = redv[tid][c]; int i2 = redi[tid][c];
      if (v > bv || (v == bv && i2 < bi)) { bv = v; bi = i2; }
    }
    j0[(size_t)b * NN + nbase + tid] = bi;
  }
}

// ------------------------------- K4: pairwise overlap score over K parents + top-16 -> knn
__global__ void __launch_bounds__(256) k4_knn(const float* __restrict__ mu_parent,
                                              const float* __restrict__ Sig_parent,
                                              const float* __restrict__ mask_parent,
                                              int* __restrict__ knn) {
  __shared__ float sc[KK];
  __shared__ float sv[256];
  __shared__ int   si[256];
  int tid = threadIdx.x;
  int b = blockIdx.x / KK, i = blockIdx.x % KK;
  const float* mui = mu_parent + ((size_t)b * KK + i) * 3;
  const float* Si  = Sig_parent + ((size_t)b * KK + i) * 9;
  float mi0 = mui[0], mi1 = mui[1], mi2 = mui[2];
  float SiL[9];
#pragma unroll
  for (int t = 0; t < 9; ++t) SiL[t] = Si[t];
  float mski = mask_parent[(size_t)b * KK + i];
  for (int j = tid; j < KK; j += 256) {
    const float* muj = mu_parent + ((size_t)b * KK + j) * 3;
    const float* Sj  = Sig_parent + ((size_t)b * KK + j) * 9;
    float d0 = mi0 - muj[0], d1 = mi1 - muj[1], d2 = mi2 - muj[2];
    float M00 = SiL[0] + Sj[0] + 1e-6f, M01 = SiL[1] + Sj[1],         M02 = SiL[2] + Sj[2];
    float M10 = SiL[3] + Sj[3],         M11 = SiL[4] + Sj[4] + 1e-6f, M12 = SiL[5] + Sj[5];
    float M20 = SiL[6] + Sj[6],         M21 = SiL[7] + Sj[7],         M22 = SiL[8] + Sj[8] + 1e-6f;
    float c00 = M11 * M22 - M12 * M21;
    float c01 = M12 * M20 - M10 * M22;
    float c02 = M10 * M21 - M11 * M20;
    float det = M00 * c00 + M01 * c01 + M02 * c02;
    float i01 = M02 * M21 - M01 * M22, i02 = M01 * M12 - M02 * M11;
    float i11 = M00 * M22 - M02 * M20, i12 = M02 * M10 - M00 * M12;
    float i21 = M01 * M20 - M00 * M21, i22 = M00 * M11 - M01 * M10;
    float invd = 1.0f / det;
    float x0 = (c00 * d0 + i01 * d1 + i02 * d2) * invd;
    float x1 = (c01 * d0 + i11 * d1 + i12 * d2) * invd;
    float x2 = (c02 * d0 + i21 * d1 + i22 * d2) * invd;
    float score = -0.5f * (d0 * x0 + d1 * x1 + d2 * x2) - 0.5f * logf(det);
    if (mski * mask_parent[(size_t)b * KK + j] < 0.5f) score = NEG_BIG;
    sc[j] = score;
  }
  __syncthreads();
  for (int r = 0; r < RR; ++r) {
    float bv = -__builtin_inff(); int bi = KK;
    for (int j = tid; j < KK; j += 256) {
      float v = sc[j];
      if (v > bv || (v == bv && j < bi)) { bv = v; bi = j; }
    }
    sv[tid] = bv; si[tid] = bi;
    __syncthreads();
    for (int off = 128; off > 0; off >>= 1) {
      if (tid < off) {
        float v = sv[tid + off]; int i2 = si[tid + off];
        if (v > sv[tid] || (v == sv[tid] && i2 < si[tid])) { sv[tid] = v; si[tid] = i2; }
      }
      __syncthreads();
    }
    if (tid == 0) { knn[((size_t)b * KK + i) * RR + r] = si[0]; sc[si[0]] = -__builtin_inff(); }
    __syncthreads();
  }
}

// -------------------------- K5: per-parent Sigma^-1 and logdet (reused across all nodes)
__global__ void __launch_bounds__(256) k5_siginv(const float* __restrict__ Sig_parent,
                                                 float* __restrict__ siginv,
                                                 float* __restrict__ slogdet) {
  int idx = blockIdx.x * 256 + threadIdx.x;
  if (idx >= BB * KK) return;
  const float* S = Sig_parent + (size_t)idx * 9;
  float M00 = S[0], M01 = S[1], M02 = S[2];
  float M10 = S[3], M11 = S[4], M12 = S[5];
  float M20 = S[6], M21 = S[7], M22 = S[8];
  float c00 = M11 * M22 - M12 * M21;
  float c01 = M12 * M20 - M10 * M22;
  float c02 = M10 * M21 - M11 * M20;
  float det = M00 * c00 + M01 * c01 + M02 * c02;
  float invd = 1.0f / det;
  float* o = siginv + (size_t)idx * 9;
  o[0] = c00 * invd;                      o[1] = (M02 * M21 - M01 * M22) * invd; o[2] = (M01 * M12 - M02 * M11) * invd;
  o[3] = c01 * invd;                      o[4] = (M00 * M22 - M02 * M20) * invd; o[5] = (M02 * M10 - M00 * M12) * invd;
  o[6] = c02 * invd;                      o[7] = (M01 * M20 - M00 * M21) * invd; o[8] = (M00 * M11 - M01 * M10) * invd;
  slogdet[idx] = logf(det);
}

// ------------------------- K6: gathered 16-way attention + moments + scores + occ/loss
__global__ void __launch_bounds__(256) k6_attend(const float* __restrict__ q,
                                                 const float* __restrict__ kmat,
                                                 const float* __restrict__ vmat,
                                                 const float* __restrict__ mu_parent,
                                                 const float* __restrict__ Sig_parent,
                                                 const float* __restrict__ siginv,
                                                 const float* __restrict__ slogdet,
                                                 const float* __restrict__ node_mask,
                                                 const int* __restrict__ j0w,
                                                 const int* __restrict__ knn,
                                                 float* __restrict__ s0_out,
                                                 float* __restrict__ mu0_out,
                                                 float* __restrict__ Sig0_out,
                                                 float* __restrict__ Bl_out,
                                                 float* __restrict__ occ,
                                                 float* __restrict__ attach_acc) {
  int wave = threadIdx.x >> 5, lane = threadIdx.x & 31;
  int node = blockIdx.x * 8 + wave;
  int b = node >> 12;
  float m = node_mask[node];
  int jj = j0w[node];
  const int* krowi = knn + ((size_t)b * KK + jj) * RR;
  int pidx[RR];
#pragma unroll
  for (int r = 0; r < RR; ++r) pidx[r] = krowi[r];
  float qreg[12];
  const float* qrow = q + (size_t)node * CC;
#pragma unroll
  for (int i = 0; i < 12; ++i) qreg[i] = qrow[lane + 32 * i];
  const float inv_sqrt_c = 0.05103103630798288f;
  float w[RR];
#pragma unroll
  for (int r = 0; r < RR; ++r) {
    const float* krow = kmat + ((size_t)b * KK + pidx[r]) * CC;
    float acc = 0.f;
#pragma unroll
    for (int i = 0; i < 12; ++i) acc += qreg[i] * krow[lane + 32 * i];
    w[r] = wave_sum32(acc) * inv_sqrt_c;
  }
  // softmax over R (constant mask bias is shift-invariant -> omitted)
  float mx = w[0];
#pragma unroll
  for (int r = 1; r < RR; ++r) mx = fmaxf(mx, w[r]);
  float se = 0.f;
#pragma unroll
  for (int r = 0; r < RR; ++r) { w[r] = expf(w[r] - mx); se += w[r]; }
  float ise = 1.f / se;
#pragma unroll
  for (int r = 0; r < RR; ++r) w[r] *= ise;
  if (lane < RR) Bl_out[(size_t)node * RR + lane] = w[lane];
  // s0 = sum_r w_r * v[p_r]
#pragma unroll
  for (int i = 0; i < 12; ++i) {
    int c = lane + 32 * i;
    float acc = 0.f;
#pragma unroll
    for (int r = 0; r < RR; ++r) acc += w[r] * vmat[((size_t)b * KK + pidx[r]) * CC + c];
    s0_out[(size_t)node * CC + c] = acc * m;
  }
  // mu0, Sig0, attach score (tiny 3x3 math, redundant per lane)
  float mu0x = 0.f, mu0y = 0.f, mu0z = 0.f;
#pragma unroll
  for (int r = 0; r < RR; ++r) {
    const float* mp = mu_parent + ((size_t)b * KK + pidx[r]) * 3;
    mu0x += w[r] * mp[0]; mu0y += w[r] * mp[1]; mu0z += w[r] * mp[2];
  }
  float S0[9] = {0, 0, 0, 0, 0, 0, 0, 0, 0};
  float att = 0.f;
#pragma unroll
  for (int r = 0; r < RR; ++r) {
    size_t pi = (size_t)b * KK + pidx[r];
    const float* mp = mu_parent + pi * 3;
    float d0 = mp[0] - mu0x, d1 = mp[1] - mu0y, d2 = mp[2] - mu0z;
    const float* Sg = Sig_parent + pi * 9;
    S0[0] += w[r] * (Sg[0] + d0 * d0); S0[1] += w[r] * (Sg[1] + d0 * d1); S0[2] += w[r] * (Sg[2] + d0 * d2);
    S0[3] += w[r] * (Sg[3] + d1 * d0); S0[4] += w[r] * (Sg[4] + d1 * d1); S0[5] += w[r] * (Sg[5] + d1 * d2);
    S0[6] += w[r] * (Sg[6] + d2 * d0); S0[7] += w[r] * (Sg[7] + d2 * d1); S0[8] += w[r] * (Sg[8] + d2 * d2);
    const float* Iv = siginv + pi * 9;
    float x0 = Iv[0] * d0 + Iv[1] * d1 + Iv[2] * d2;
    float x1 = Iv[3] * d0 + Iv[4] * d1 + Iv[5] * d2;
    float x2 = Iv[6] * d0 + Iv[7] * d1 + Iv[8] * d2;
    float maha = d0 * x0 + d1 * x1 + d2 * x2;
    att += w[r] * (-0.5f * maha - 0.5f * slogdet[pi]);
  }
  S0[0] += JITTER_; S0[4] += JITTER_; S0[8] += JITTER_;
  if (lane < 3) mu0_out[(size_t)node * 3 + lane] = (lane == 0) ? mu0x : (lane == 1 ? mu0y : mu0z);
  if (lane < 9) Sig0_out[(size_t)node * 9 + lane] = S0[lane];
  if (lane == 0) atomicAdd(attach_acc, m * att);
  if (lane < RR) atomicAdd(&occ[(size_t)b * KK + pidx[lane]], w[lane] * m);
}

// ----------------------------------------- K7: occ normalize + loss
__global__ void __launch_bounds__(256) k7_finalize(const float* __restrict__ occ,
                                                   const float* __restrict__ attach_acc,
                                                   const float* __restrict__ scal,
                                                   float* __restrict__ occn_out,
                                                   float* __restrict__ loss_out) {
  __shared__ float red[256];
  __shared__ float sums[BB];
  int tid = threadIdx.x;
  for (int b = 0; b < BB; ++b) {
    float s = occ[b * KK + tid] + occ[b * KK + tid + 256];
    red[tid] = s; __syncthreads();
    for (int off = 128; off > 0; off >>= 1) { if (tid < off) red[tid] += red[tid + off]; __syncthreads(); }
    if (tid == 0) sums[b] = fmaxf(red[0], 1e-9f);
    __syncthreads();
  }
  for (int idx = tid; idx < BB * KK; idx += 256)
    occn_out[idx] = occ[idx] / sums[idx >> 9];
  if (tid == 0) {
    float dnm = fmaxf(scal[2], 1.0f);
    loss_out[0] = -attach_acc[0] / dnm;   // W_ATTACH=1, W_ENTB=0
  }
}

// ================================================================ launcher
extern "C" void kernel_launch(void* const* d_in, const int* in_sizes, int n_in,
                              void* d_out, int out_size, void* d_ws, size_t ws_size,
                              hipStream_t stream) {
  const float* s_parent    = (const float*)d_in[0];
  const float* mu_parent   = (const float*)d_in[1];
  const float* Sig_parent  = (const float*)d_in[2];
  const float* mask_parent = (const float*)d_in[3];
  const float* node_mask   = (const float*)d_in[4];
  const int*   res_idx     = (const int*)d_in[5];
  const float* Bmat        = (const float*)d_in[6];
  const float* Wq          = (const float*)d_in[7];
  const float* Wk          = (const float*)d_in[8];
  const float* Wv          = (const float*)d_in[9];
  const float* ln_g        = (const float*)d_in[10];
  const float* ln_bb       = (const float*)d_in[11];

  float* ws = (float*)d_ws;
  float* q0    = ws;
  float* qm    = q0 + (size_t)BB * NN * CC;
  float* kmat  = qm + (size_t)BB * NN * CC;
  float* vmat  = kmat + (size_t)BB * KK * CC;
  int*   j0w   = (int*)(vmat + (size_t)BB * KK * CC);
  int*   knn   = j0w + BB * NN;
  float* sinv  = (float*)(knn + BB * KK * RR);
  float* sld   = sinv + (size_t)BB * KK * 9;
  float* occ   = sld + BB * KK;
  float* attach= occ + BB * KK;
  float* scal  = attach + 1;          // [0],[1]=denom per batch, [2]=mask total

  float* out      = (float*)d_out;
  float* s0_out   = out;
  float* mu0_out  = s0_out + (size_t)BB * NN * CC;
  float* Sig0_out = mu0_out + (size_t)BB * NN * 3;
  float* Bl_out   = Sig0_out + (size_t)BB * NN * 9;
  float* occn_out = Bl_out + (size_t)BB * NN * RR;
  float* loss_out = occn_out + (size_t)BB * KK;

  hipMemsetAsync(occ, 0, (BB * KK + 1) * sizeof(float), stream);   // occ + attach accumulator

  k0_mask_sums<<<1, 256, 0, stream>>>(node_mask, scal);
  k1_embed_ln<<<BB * NN / 8, 256, 0, stream>>>(res_idx, node_mask, Bmat, ln_g, ln_bb, scal, q0);

  dim3 gq(BB * NN / 16, CC / 64);
  dim3 gk(BB * KK / 16, CC / 64);
  k2_gemm_wmma<<<gq, 128, 0, stream>>>(q0, Wq, qm);          // q  = LN(emb) @ Wq
  k2_gemm_wmma<<<gk, 128, 0, stream>>>(s_parent, Wk, kmat);  // k  = s @ Wk
  k2_gemm_wmma<<<gk, 128, 0, stream>>>(s_parent, Wv, vmat);  // v  = s @ Wv

  k3_logits_argmax<<<BB * NN / 16, 128, 0, stream>>>(qm, kmat, mask_parent, j0w);
  k4_knn<<<BB * KK, 256, 0, stream>>>(mu_parent, Sig_parent, mask_parent, knn);
  k5_siginv<<<(BB * KK + 255) / 256, 256, 0, stream>>>(Sig_parent, sinv, sld);
  k6_attend<<<BB * NN / 8, 256, 0, stream>>>(qm, kmat, vmat, mu_parent, Sig_parent, sinv, sld,
                                             node_mask, j0w, knn,
                                             s0_out, mu0_out, Sig0_out, Bl_out, occ, attach);
  k7_finalize<<<1, 256, 0, stream>>>(occ, attach, scal, occn_out, loss_out);
}